// QLayer_180388626716
// MI455X (gfx1250) — compile-verified
//
#include <hip/hip_runtime.h>

typedef __attribute__((ext_vector_type(4))) float v4f;
typedef __attribute__((ext_vector_type(2))) float v2f;
typedef __attribute__((ext_vector_type(8))) float v8f;

#define BINS 16
#define BLOCK 256
#define WAVES_PER_BLOCK (BLOCK / 32)

// ---------------------------------------------------------------------------
// Kernel 1: stream quantize x -> xq, accumulate 16-bin histogram.
//   - v4f (b128) non-temporal loads/stores: pure streaming, no reuse.
//   - Markstein correctly-rounded division: y1 = RN(1/s) once, then
//     q0 = x*y1; r = fma(-s,q0,x); q = fma(r,y1,q0)  == RN(x/s).
//   - Wave32 conflict-free LDS histogram: hist[wave][bin][lane]; lane l only
//     touches banks {l, l+32}, and owns its column exclusively within a wave.
//   - Final 16x16 partial-count matrix is reduced to per-bin totals by wave 0
//     with 4 accumulating V_WMMA_F32_16X16X4_F32 ops (D = ones(16x4)*P + C).
// ---------------------------------------------------------------------------
__global__ __launch_bounds__(BLOCK) void quant_hist_kernel(
    const float* __restrict__ x, const float* __restrict__ sp,
    float* __restrict__ xq, unsigned* __restrict__ ghist, int n)
{
    __shared__ unsigned hist[WAVES_PER_BLOCK * BINS * 32];
    __shared__ float    partialf[16 * BINS];   // P[r][bin], r = lane-pair group

    const int tid  = threadIdx.x;
    const int lane = tid & 31;
    const int wave = tid >> 5;

    for (int i = tid; i < WAVES_PER_BLOCK * BINS * 32; i += BLOCK) hist[i] = 0u;
    __syncthreads();

    const float s  = sp[0];
    const float y1 = 1.0f / s;   // correctly-rounded reciprocal (uniform, once)

    unsigned* whist = &hist[wave * BINS * 32];

    const int gtid    = blockIdx.x * BLOCK + tid;
    const int gstride = gridDim.x * BLOCK;
    const int nv      = n >> 2;

    const v4f* __restrict__ xv = (const v4f*)x;
    v4f* __restrict__ qv       = (v4f*)xq;

    for (int i = gtid; i < nv; i += gstride) {
        v4f xx = __builtin_nontemporal_load(&xv[i]);
        v4f outv;
#pragma unroll
        for (int j = 0; j < 4; ++j) {
            float xs = xx[j];
            float q0 = xs * y1;
            float r  = fmaf(-s, q0, xs);
            float q  = fmaf(r, y1, q0);          // RN(xs / s)
            q        = fminf(fmaxf(q, -8.0f), 7.0f);
            float b  = rintf(q);                 // round-half-even == jnp.round
            outv[j]  = b * s;
            int ib   = (int)b + 8;               // 0..15
            atomicAdd(&whist[ib * 32 + lane], 1u);
        }
        __builtin_nontemporal_store(outv, &qv[i]);
    }

    // scalar tail (n not multiple of 4)
    for (int i = (nv << 2) + gtid; i < n; i += gstride) {
        float xs = x[i];
        float q0 = xs * y1;
        float r  = fmaf(-s, q0, xs);
        float q  = fmaf(r, y1, q0);
        q        = fminf(fmaxf(q, -8.0f), 7.0f);
        float b  = rintf(q);
        xq[i]    = b * s;
        int ib   = (int)b + 8;
        atomicAdd(&whist[ib * 32 + lane], 1u);
    }

    __syncthreads();

    // Stage 1: 256 threads fold 8 waves x 32 lanes -> P[16][16] (exact ints in f32).
    {
        const int bin = tid & 15;
        const int sub = tid >> 4;     // 0..15, each covers one lane pair
        unsigned ssum = 0;
#pragma unroll
        for (int w = 0; w < WAVES_PER_BLOCK; ++w) {
            const unsigned* hb = &hist[(w * BINS + bin) * 32 + (sub << 1)];
            ssum += hb[0] + hb[1];
        }
        partialf[sub * 16 + bin] = (float)ssum;  // per-block counts <= 2^24: exact
    }
    __syncthreads();

    // Stage 2: wave 0 reduces P over rows with the matrix pipe.
    // D = ones(16x4) . P[4g..4g+3][:] accumulated over g=0..3
    //   => every row of D holds totals[bin]; read row M=0 from acc[0].
    if (wave == 0) {
        v2f a;  a[0] = 1.0f; a[1] = 1.0f;        // A = ones (16x4)
        v8f acc = {};
#pragma unroll
        for (int g = 0; g < 4; ++g) {
            // B (4x16): VGPR0 = {K=4g+0 | lanes 0-15, K=4g+2 | lanes 16-31},
            //           VGPR1 = {K=4g+1,            K=4g+3}
            int row0 = (g << 2) + ((lane >> 4) << 1);
            int col  = lane & 15;
            v2f b;
            b[0] = partialf[row0 * 16 + col];
            b[1] = partialf[(row0 + 1) * 16 + col];
            acc = __builtin_amdgcn_wmma_f32_16x16x4_f32(
                false, a, false, b, (short)0, acc, false, false);
        }
        if (lane < 16) {
            atomicAdd(&ghist[lane], (unsigned)acc[0]);   // exact integer in f32
        }
    }
}

// ---------------------------------------------------------------------------
// Kernel 2: finalize the loss from bin counts.
// Every element in bin k equals v_k = (k-8)*s exactly, so the reference's
// segment sums are sm = cnt*v, sq = cnt*v^2 (up to its own fp32 noise).
// ---------------------------------------------------------------------------
__global__ void finalize_loss_kernel(const unsigned* __restrict__ ghist,
                                     const float* __restrict__ sp,
                                     float* __restrict__ out, int n)
{
    if (threadIdx.x != 0 || blockIdx.x != 0) return;
    const float s = sp[0];
    double loss = 0.0;
#pragma unroll
    for (int k = 0; k < BINS - 1; ++k) {         // last bin dropped
        double cnt = (double)ghist[k];
        float  v   = (float)(k - 8) * s;         // bin value (bit-exact xq)
        float  c   = -8.0f + s * ((float)k + 0.5f);
        double sm  = cnt * (double)v;
        double sq  = cnt * (double)v * (double)v;
        double safec = cnt > 1.0 ? cnt : 1.0;    // max(cnt, 1)
        double mse = (sq - 2.0 * (double)c * sm + (double)c * (double)c * cnt) / safec;
        if (!(cnt > 0.0)) mse = 0.0;
        double var = (sq - sm * sm / safec) / ((cnt - 1.0) > 1.0 ? (cnt - 1.0) : 1.0);
        if (!(cnt > 1.0)) var = 0.0;             // (exactly 0 analytically)
        loss += mse + var;
    }
    out[n] = (float)loss;
}

// ---------------------------------------------------------------------------
extern "C" void kernel_launch(void* const* d_in, const int* in_sizes, int n_in,
                              void* d_out, int out_size, void* d_ws, size_t ws_size,
                              hipStream_t stream)
{
    (void)n_in; (void)out_size; (void)ws_size;
    const float* x = (const float*)d_in[0];
    const float* s = (const float*)d_in[1];
    float* out     = (float*)d_out;
    const int n    = in_sizes[0];

    unsigned* ghist = (unsigned*)d_ws;
    hipMemsetAsync(ghist, 0, BINS * sizeof(unsigned), stream);

    int nv = n >> 2;
    int blocks = (nv + BLOCK - 1) / BLOCK;
    if (blocks > 4096) blocks = 4096;
    if (blocks < 1) blocks = 1;

    quant_hist_kernel<<<blocks, BLOCK, 0, stream>>>(x, s, out, ghist, n);
    finalize_loss_kernel<<<1, 32, 0, stream>>>(ghist, s, out, n);
}